// HierarchicalMemoryRouter_24421184045397
// MI455X (gfx1250) — compile-verified
//
#include <hip/hip_runtime.h>

typedef __attribute__((ext_vector_type(2))) float v2f;
typedef __attribute__((ext_vector_type(8))) float v8f;

#define D        1024
#define SLOTS    12      // NSSM + NMSM router outputs
#define NSSM     8
#define NLSM     16
#define K_TILE   64
#define NCHUNK   (D / K_TILE)
#define LDA      68      // padded LDS stride (dwords): conflict-free fragment reads, 16B aligned
#define ROWS_PER_BLOCK 128

// Output flat offsets (reference return tuple, concatenated)
#define OFF_WEIGHTED 0        // [12,1024]
#define OFF_SSM3     12288    // [8,1024]
#define OFF_AGE3     20480    // [8]
#define OFF_LKEYS    20488    // [16,1024]
#define OFF_LVALS    36872    // [16,1024]
#define OFF_LAGE     53256    // [16]
#define OFF_SIMMAX   53272    // [1]

// ---------------------------------------------------------------------------
// Kernel 0: zero the avg_weights accumulator in workspace (graph-replay safe)
// ---------------------------------------------------------------------------
__global__ void zero_ws_kernel(float* __restrict__ ws) {
    if (threadIdx.x < 16) ws[threadIdx.x] = 0.0f;
}

// ---------------------------------------------------------------------------
// Kernel 1: streaming pass over input_stream (memory-bound, 512 MB @ 23.3 TB/s).
//   logits = input @ router_w.T + router_b  (N=12 padded to 16)
//   per-row softmax over 12, accumulate column sums into ws_acc[12].
// One wave = one 16x16 logits tile via V_WMMA_F32_16X16X4_F32.
// Global->LDS staging uses GLOBAL_LOAD_ASYNC_TO_LDS_B128 (ASYNCcnt) with
// double-buffered LDS so the DMA of chunk c+1 overlaps the WMMAs on chunk c.
// ---------------------------------------------------------------------------
__global__ __launch_bounds__(256) void router_softmax_pass(
    const float* __restrict__ input,
    const float* __restrict__ rw,      // [12, D]
    const float* __restrict__ rb,      // [12]
    float* __restrict__ ws_acc)        // [12] global accumulator
{
    __shared__ float lds_a[2][ROWS_PER_BLOCK * LDA]; // ping-pong A tiles (128 x 64)
    __shared__ float lds_b[2][16 * LDA];             // ping-pong B tiles (16 x 64, padded)
    __shared__ float lds_avg[16];

    const int tid  = threadIdx.x;
    const int lane = tid & 31;
    const int wave = tid >> 5;
    const int m    = lane & 15;   // A: row within tile / B,C: column N
    const int kh   = lane >> 4;   // K-half select (lanes 16-31 hold K+2)
    const int r0   = wave * 16;   // this wave's first row in the LDS tile
    const long base = (long)blockIdx.x * ROWS_PER_BLOCK;

    // One-time zero of B pad rows 12..15 (never overwritten by the DMA loads)
    if (tid < 128) {
        int buf = tid >> 6;
        int row = 12 + ((tid >> 4) & 3);
        int cc  = tid & 15;
        *(float4*)&lds_b[buf][row * LDA + cc * 4] = make_float4(0.f, 0.f, 0.f, 0.f);
    }
    if (tid < 16) lds_avg[tid] = 0.0f;

    // Per-thread async-copy geometry: thread owns float4 column c4 of rows rB+16i
    const int rB = tid >> 4;      // 0..15
    const int c4 = tid & 15;      // 0..15
    unsigned ldsA[2], ldsB[2];
    ldsA[0] = (unsigned)(size_t)&lds_a[0][rB * LDA + c4 * 4];
    ldsA[1] = (unsigned)(size_t)&lds_a[1][rB * LDA + c4 * 4];
    ldsB[0] = (unsigned)(size_t)&lds_b[0][rB * LDA + c4 * 4];
    ldsB[1] = (unsigned)(size_t)&lds_b[1][rB * LDA + c4 * 4];
    const float* gA = input + (base + rB) * (long)D + c4 * 4;
    const float* gB = rw + (long)rB * D + c4 * 4;   // valid only when rB < SLOTS

    auto stage = [&](int buf, int k0) {
        #pragma unroll
        for (int i = 0; i < 8; ++i) {
            unsigned l = ldsA[buf] + (unsigned)(i * 16 * LDA * 4);
            const float* g = gA + k0 + (long)i * 16 * D;
            asm volatile("global_load_async_to_lds_b128 %0, %1, off"
                         :: "v"(l), "v"(g) : "memory");
        }
        if (rB < SLOTS) {
            const float* g = gB + k0;
            asm volatile("global_load_async_to_lds_b128 %0, %1, off"
                         :: "v"(ldsB[buf]), "v"(g) : "memory");
        }
    };

    // Prologue: fill buffer 0
    stage(0, 0);
    asm volatile("s_wait_asynccnt 0x0" ::: "memory");
    __syncthreads();

    v8f acc = {}; // 16x16 fp32 accumulator (8 VGPRs)

    for (int c = 0; c < NCHUNK; ++c) {
        const int buf = c & 1;
        if (c + 1 < NCHUNK) stage(buf ^ 1, (c + 1) * K_TILE);   // overlap DMA w/ math

        const float* A = &lds_a[buf][(r0 + m) * LDA];
        const float* B = &lds_b[buf][m * LDA];
        #pragma unroll
        for (int ks = 0; ks < K_TILE / 4; ++ks) {
            int kl = ks * 4 + 2 * kh;
            v2f a = *(const v2f*)&A[kl];   // (row m, K=kl..kl+1)
            v2f b = *(const v2f*)&B[kl];   // (K=kl..kl+1, col m)
            acc = __builtin_amdgcn_wmma_f32_16x16x4_f32(
                /*neg_a=*/false, a, /*neg_b=*/false, b,
                /*c_mod=*/(short)0, acc, /*reuse_a=*/false, /*reuse_b=*/false);
        }
        asm volatile("s_wait_asynccnt 0x0" ::: "memory");
        __syncthreads();
    }

    // ---- per-row softmax (rows live across 16-lane halves; VGPR j = rows j / j+8)
    const float bias = (m < SLOTS) ? rb[m] : 0.0f;
    float psum = 0.0f; // per-lane: sum over 8 rows of softmax prob for column m
    #pragma unroll
    for (int j = 0; j < 8; ++j) {
        float x = (m < SLOTS) ? (acc[j] + bias) : -3.4e38f;
        float mx = x;
        #pragma unroll
        for (int off = 1; off < 16; off <<= 1)
            mx = fmaxf(mx, __shfl_xor(mx, off, 16));
        float e = (m < SLOTS) ? __expf(x - mx) : 0.0f;
        float s = e;
        #pragma unroll
        for (int off = 1; off < 16; off <<= 1)
            s += __shfl_xor(s, off, 16);
        psum += e / s;
    }
    // combine the two halves (rows 0-7 with rows 8-15) for this column
    psum += __shfl_xor(psum, 16, 32);

    if (lane < SLOTS) atomicAdd(&lds_avg[lane], psum);
    __syncthreads();
    if (tid < SLOTS) atomicAdd(&ws_acc[tid], lds_avg[tid]);
}

// ---------------------------------------------------------------------------
// Kernel 2: everything else (single block, 1024 threads = 32 waves)
// ---------------------------------------------------------------------------
__global__ __launch_bounds__(1024) void finalize_kernel(
    const float* __restrict__ input,
    const float* __restrict__ ssm,      // [8, D]
    const float* __restrict__ msm,      // [4, D]
    const float* __restrict__ cw,       // [D, D]
    const float* __restrict__ cb,       // [D]
    const float* __restrict__ sage,     // [8]
    const float* __restrict__ lkeys,    // [16, D]
    const float* __restrict__ lvals,    // [16, D]
    const float* __restrict__ lage,     // [16]
    const float* __restrict__ ws_acc,   // [12] softmax column sums
    float* __restrict__ out, int T)
{
    __shared__ float rm[D];       // recent_mean
    __shared__ float tgt[D];      // compress(recent_mean)
    __shared__ float aw[16];      // avg_weights
    __shared__ float dotk[NLSM], nk2[NLSM];
    __shared__ float sc[4];       // 0:norm_m 1:sim_max 2:cond_sim 3:cond_age
    __shared__ int   si[4];       // 0:victim_ssm 1:sim_idx 2:victim_lsm

    const int tid  = threadIdx.x;     // == d index (D == blockDim.x)
    const int lane = tid & 31;
    const int wave = tid >> 5;

    // recent_mean over last 8 rows
    {
        float s = 0.0f;
        #pragma unroll
        for (int i = 0; i < 8; ++i) s += input[(long)(T - 8 + i) * D + tid];
        rm[tid] = s * 0.125f;
    }
    if (tid < SLOTS) aw[tid] = ws_acc[tid] / (float)T;
    __syncthreads();

    // weighted = slots * avg_weights[:, None]
    #pragma unroll
    for (int s = 0; s < SLOTS; ++s) {
        float slot = (s < NSSM) ? ssm[s * D + tid] : msm[(s - NSSM) * D + tid];
        out[OFF_WEIGHTED + s * D + tid] = slot * aw[s];
    }

    // target = compress_w @ recent_mean + compress_b; one wave per output row
    for (int d = wave; d < D; d += 32) {
        float p = 0.0f;
        for (int k = lane; k < D; k += 32) p += cw[d * D + k] * rm[k];
        #pragma unroll
        for (int off = 16; off > 0; off >>= 1) p += __shfl_xor(p, off, 32);
        if (lane == 0) tgt[d] = p + cb[d];
    }

    // lsm key dots + norms (waves 0..15), |recent_mean| (wave 16)
    if (wave < NLSM) {
        float pd = 0.0f, pn = 0.0f;
        for (int k = lane; k < D; k += 32) {
            float kv = lkeys[wave * D + k];
            pd += kv * rm[k];
            pn += kv * kv;
        }
        #pragma unroll
        for (int off = 16; off > 0; off >>= 1) {
            pd += __shfl_xor(pd, off, 32);
            pn += __shfl_xor(pn, off, 32);
        }
        if (lane == 0) { dotk[wave] = pd; nk2[wave] = pn; }
    } else if (wave == 16) {
        float p = 0.0f;
        for (int k = lane; k < D; k += 32) p += rm[k] * rm[k];
        #pragma unroll
        for (int off = 16; off > 0; off >>= 1) p += __shfl_xor(p, off, 32);
        if (lane == 0) sc[0] = sqrtf(p);
    }
    __syncthreads();

    if (tid == 0) {
        // victim_ssm = argmin(avg_weights[:8]) (first min)
        int v = 0; float mv = aw[0];
        for (int i = 1; i < NSSM; ++i) if (aw[i] < mv) { mv = aw[i]; v = i; }
        si[0] = v;
        // cond_age = max(ssm_age + 1) > 50
        float maxage = -3.4e38f;
        for (int i = 0; i < NSSM; ++i) maxage = fmaxf(maxage, sage[i] + 1.0f);
        sc[3] = (maxage > 50.0f) ? 1.0f : 0.0f;
        // cosine sims; sim_idx = argmax (first max)
        float nm = sc[0];
        int bi = 0; float bs = -3.4e38f;
        for (int j = 0; j < NLSM; ++j) {
            float s = dotk[j] / fmaxf(sqrtf(nk2[j]) * nm, 1e-8f);
            if (s > bs) { bs = s; bi = j; }
        }
        si[1] = bi; sc[1] = bs;
        sc[2] = (bs >= 0.8f) ? 1.0f : 0.0f;
        // victim_lsm = argmax(lsm_age + 1) (first max)
        int vl = 0; float ma = lage[0] + 1.0f;
        for (int j = 1; j < NLSM; ++j) {
            float a = lage[j] + 1.0f;
            if (a > ma) { ma = a; vl = j; }
        }
        si[2] = vl;
        out[OFF_SIMMAX] = bs;
    }
    __syncthreads();

    const int v = si[0], simi = si[1], vl = si[2];
    const bool cage = sc[3] > 0.5f, csim = sc[2] > 0.5f;
    const float nm = sc[0];

    // ssm3 rows (compose the up-to-three sequential updates)
    #pragma unroll
    for (int i = 0; i < NSSM; ++i) {
        float val = ssm[i * D + tid] * 0.9f + 0.1f * tgt[tid];        // ssm1
        if (i == v && cage) val = val * 0.9f + 0.1f * tgt[tid];       // ssm2 eviction
        if (i == v && csim) val = val * 0.9f + 0.1f * lvals[simi * D + tid]; // ssm3 blend
        out[OFF_SSM3 + i * D + tid] = val;
    }
    if (tid < NSSM) {
        float a = sage[tid] + 1.0f;
        if (tid == v && (cage || csim)) a = 0.0f;
        out[OFF_AGE3 + tid] = a;
    }

    // lsm keys / values (branch B overwrites victim row only when !cond_sim)
    const float inv = 1.0f / fmaxf(nm, 1e-12f);
    #pragma unroll
    for (int j = 0; j < NLSM; ++j) {
        float kv = lkeys[j * D + tid];
        float vv = lvals[j * D + tid];
        if (!csim && j == vl) { kv = rm[tid] * inv; vv = tgt[tid]; }
        out[OFF_LKEYS + j * D + tid] = kv;
        out[OFF_LVALS + j * D + tid] = vv;
    }
    if (tid < NLSM) {
        float a = lage[tid] + 1.0f;
        if (!csim && tid == vl) a = 0.0f;
        out[OFF_LAGE + tid] = a;
    }
}

// ---------------------------------------------------------------------------
extern "C" void kernel_launch(void* const* d_in, const int* in_sizes, int n_in,
                              void* d_out, int out_size, void* d_ws, size_t ws_size,
                              hipStream_t stream) {
    const float* input = (const float*)d_in[0];
    const float* ssm   = (const float*)d_in[1];
    const float* msm   = (const float*)d_in[2];
    const float* rw    = (const float*)d_in[3];
    const float* rb    = (const float*)d_in[4];
    const float* cw    = (const float*)d_in[5];
    const float* cb    = (const float*)d_in[6];
    const float* sage  = (const float*)d_in[7];
    const float* lk    = (const float*)d_in[8];
    const float* lv    = (const float*)d_in[9];
    const float* la    = (const float*)d_in[10];
    float* out = (float*)d_out;
    float* acc = (float*)d_ws;

    const int T = in_sizes[0] / D; // 131072

    zero_ws_kernel<<<dim3(1), dim3(32), 0, stream>>>(acc);
    router_softmax_pass<<<dim3(T / ROWS_PER_BLOCK), dim3(256), 0, stream>>>(input, rw, rb, acc);
    finalize_kernel<<<dim3(1), dim3(1024), 0, stream>>>(input, ssm, msm, cw, cb, sage,
                                                        lk, lv, la, acc, out, T);
}